// DiversityLoss_47184510714393
// MI455X (gfx1250) — compile-verified
//
#include <hip/hip_runtime.h>

typedef float v2f __attribute__((ext_vector_type(2)));
typedef float v8f __attribute__((ext_vector_type(8)));

#define MARGIN     0.1f
#define B_         64
#define M_         32
#define K_         16384
#define SPLIT      4
#define KS         (K_ / SPLIT)   // 4096 K per workgroup
#define KT         256            // K tile staged in LDS
#define STAGES     (KS / KT)      // 16
#define LDS_STRIDE 260            // 256 + 4 pad -> conflict-free b64 frag reads
#define NTHREADS   256
#define NWAVES     8
#define BUF_FLOATS (M_ * LDS_STRIDE)   // 8320 floats = 33,280 B per buffer

// Issue one stage (32 rows x 256 cols f32) of async DMA Global -> LDS.
// Per thread: 8 x b128, per-lane LDS scatter address. Tracked by ASYNCcnt.
__device__ __forceinline__ void async_load_stage(const float* __restrict__ Ab,
                                                 float* __restrict__ buf,
                                                 int kbase, int tid) {
    const int col4 = (tid & 63) * 4;   // float4 column within tile
    const int row0 = tid >> 6;         // 0..3
    #pragma unroll
    for (int p = 0; p < 8; ++p) {
        const int row = p * 4 + row0;
        const float* g = Ab + (size_t)row * K_ + kbase + col4;
        // LDS byte offset = low 32 bits of generic shared pointer (wave-relative)
        unsigned int l = (unsigned int)(unsigned long long)(buf + row * LDS_STRIDE + col4);
        asm volatile("global_load_async_to_lds_b128 %0, %1, off"
                     :: "v"(l), "v"(g) : "memory");
    }
}

// ---------------- Kernel A: partial Gram via f32 WMMA, async double-buffer ---
__global__ __launch_bounds__(NTHREADS)
void DiversityLoss_gram_partial(const float* __restrict__ A, float* __restrict__ ws) {
    __shared__ float lds[2 * BUF_FLOATS];   // 66,560 B (<= 320 KB WGP pool)

    const int b    = blockIdx.x;
    const int s    = blockIdx.y;
    const int tid  = threadIdx.x;
    const int lane = tid & 31;
    const int wave = tid >> 5;

    const float* __restrict__ Ab = A + (size_t)b * M_ * K_;

    v8f acc00 = {}; v8f acc01 = {}; v8f acc11 = {};

    // WMMA fragment addressing (A 16x4 f32 layout; B 4x16 is the identical
    // register image for Gram, so one fragment serves both operands).
    const int frow = lane & 15;         // row within 16-row group
    const int fk   = (lane >> 4) * 2;   // K sub-offset: +0 (lanes 0-15) / +2 (16-31)
    const int kbase0 = s * KS;

    // Prefetch stage 0 into buffer 0.
    async_load_stage(Ab, lds, kbase0, tid);

    for (int st = 0; st < STAGES; ++st) {
        float* cur = lds + (st & 1) * BUF_FLOATS;

        if (st + 1 < STAGES) {
            // Issue stage st+1 DMA into the other buffer (safe: end-of-(st-1)
            // barrier guarantees every wave finished reading it), then wait for
            // the older 8 ops (in-order completion) => stage st is in LDS.
            async_load_stage(Ab, lds + ((st + 1) & 1) * BUF_FLOATS,
                             kbase0 + (st + 1) * KT, tid);
            asm volatile("s_wait_asynccnt 0x8" ::: "memory");
        } else {
            asm volatile("s_wait_asynccnt 0x0" ::: "memory");
        }
        __syncthreads();   // all waves' stage-st data visible

        // ---- wave w owns tile K columns [w*32, w*32+32) ----
        #pragma unroll
        for (int kk = 0; kk < 8; ++kk) {
            const int k0 = wave * 32 + kk * 4;
            v2f f0 = *(const v2f*)&cur[(frow     ) * LDS_STRIDE + k0 + fk];
            v2f f1 = *(const v2f*)&cur[(frow + 16) * LDS_STRIDE + k0 + fk];
            f0 *= f0;   // S = A^2 applied at fragment read (v_pk_mul_f32)
            f1 *= f1;
            acc00 = __builtin_amdgcn_wmma_f32_16x16x4_f32(false, f0, false, f0,
                                                          (short)0, acc00, false, false);
            acc01 = __builtin_amdgcn_wmma_f32_16x16x4_f32(false, f0, false, f1,
                                                          (short)0, acc01, false, false);
            acc11 = __builtin_amdgcn_wmma_f32_16x16x4_f32(false, f1, false, f1,
                                                          (short)0, acc11, false, false);
        }
        __syncthreads();   // all reads of `cur` done before it is DMA-overwritten
    }

    // ---- cross-wave reduction of the 3 unique 16x16 tiles ----
    float* red = lds;   // reuse buffer 0 (needs 8*3*256 = 6144 floats)
    #pragma unroll
    for (int i = 0; i < 8; ++i) {
        red[(wave * 3 + 0) * 256 + lane * 8 + i] = acc00[i];
        red[(wave * 3 + 1) * 256 + lane * 8 + i] = acc01[i];
        red[(wave * 3 + 2) * 256 + lane * 8 + i] = acc11[i];
    }
    __syncthreads();

    float* out = ws + ((size_t)b * SPLIT + s) * 768;
    #pragma unroll
    for (int t = 0; t < 3; ++t) {
        float sum = 0.f;
        #pragma unroll
        for (int w = 0; w < NWAVES; ++w)
            sum += red[(w * 3 + t) * 256 + tid];
        out[t * 256 + tid] = sum;   // coalesced
    }
}

// ---------------- Kernel B: combine K-splits, hinge loss, reduce -------------
__global__ __launch_bounds__(256)
void DiversityLoss_finalize(const float* __restrict__ ws, float* __restrict__ out) {
    __shared__ float red[256];
    const int tid = threadIdx.x;
    float local = 0.f;

    for (int idx = tid; idx < B_ * 768; idx += 256) {
        const int b = idx / 768;
        const int r = idx - b * 768;
        const int t = r >> 8;           // tile: 0=T00, 1=T01, 2=T11
        const int e = r & 255;          // element within tile

        float g = 0.f;
        #pragma unroll
        for (int s = 0; s < SPLIT; ++s)
            g += ws[((size_t)b * SPLIT + s) * 768 + r];

        // decode WMMA C layout: e = lane*8 + vgpr ; M = vgpr + 8*(lane>=16), N = lane&15
        const int ln = e >> 3;
        const int vr = e & 7;
        int i = vr + ((ln >> 4) << 3);
        int j = ln & 15;
        if (t == 1)      { j += 16; }
        else if (t == 2) { i += 16; j += 16; }

        if (i < j) {
            const float v = MARGIN - g;
            local += (v > 0.f) ? v : 0.f;
        }
    }

    red[tid] = local;
    __syncthreads();
    #pragma unroll
    for (int off = 128; off > 0; off >>= 1) {
        if (tid < off) red[tid] += red[tid + off];
        __syncthreads();
    }
    if (tid == 0)
        out[0] = red[0] / (496.f * 64.f);   // / count / B
}

extern "C" void kernel_launch(void* const* d_in, const int* in_sizes, int n_in,
                              void* d_out, int out_size, void* d_ws, size_t ws_size,
                              hipStream_t stream) {
    const float* A  = (const float*)d_in[0];
    float* ws       = (float*)d_ws;      // needs 64*4*768*4 = 786,432 bytes
    float* out      = (float*)d_out;

    dim3 grid(B_, SPLIT);
    DiversityLoss_gram_partial<<<grid, NTHREADS, 0, stream>>>(A, ws);
    DiversityLoss_finalize<<<1, 256, 0, stream>>>(ws, out);
}